// TrajectoryFeatureExtractor_32830730010722
// MI455X (gfx1250) — compile-verified
//
#include <hip/hip_runtime.h>
#include <hip/hip_bf16.h>
#include <math.h>

typedef __attribute__((ext_vector_type(2))) float v2f;
typedef __attribute__((ext_vector_type(8))) float v8f;

constexpr int kS   = 48;    // padded trajectory length
constexpr int kMD  = 10;    // coord dims
constexpr int kSMD = kS * kMD;      // 480 floats per trajectory
constexpr int kLST = kSMD + 1;      // 481: LDS stride (odd mod 64 -> conflict-free)
constexpr int kH   = 48;    // hidden dim
constexpr int kF   = 64;    // padded feature count (60 real + 4 zero)

// ---------------------------------------------------------------------------
// Kernel 1: per-trajectory feature extraction. One wave per block; one
// trajectory per lane, staged through LDS with coalesced loads.
// ---------------------------------------------------------------------------
__global__ __launch_bounds__(32) void traj_features_kernel(
    const float* __restrict__ coords, const int* __restrict__ lengths,
    float* __restrict__ feats)
{
    __shared__ float lds[32 * kLST];
    const int tid   = threadIdx.x;
    const int bbase = blockIdx.x * 32;

    // Coalesced cooperative staging: 32 lanes read consecutive floats.
    const float* src = coords + (size_t)bbase * kSMD;
    for (int idx = tid; idx < 32 * kSMD; idx += 32) {
        int traj = idx / kSMD;
        int off  = idx - traj * kSMD;
        lds[traj * kLST + off] = src[idx];
    }
    __syncthreads();

    const int    b    = bbase + tid;
    const float* C    = &lds[tid * kLST];
    float*       fout = feats + (size_t)b * kF;

    const int   n  = lengths[b];       // 4 <= n <= 48
    const int   m  = n - 1;            // >= 3
    const float nf = (float)n, mf = (float)m;

    // zero curvature pad [51..59] and tail pad [60..63]
    for (int j = 51; j < kF; ++j) fout[j] = 0.f;

    // coordinate mean/std over n*md valid entries
    float csum = 0.f, csumsq = 0.f;
    for (int s = 0; s < n; ++s) {
#pragma unroll
        for (int d = 0; d < kMD; ++d) {
            float v = C[s * kMD + d];
            csum += v; csumsq += v * v;
        }
    }

    // total displacement
    float tdsq = 0.f;
#pragma unroll
    for (int d = 0; d < kMD; ++d) {
        float diff = C[(n - 1) * kMD + d] - C[d];
        tdsq += diff * diff;
    }
    const float total_disp = (tdsq > 0.f) ? sqrtf(fmaxf(tdsq, 1e-30f)) : 0.f;

    const int halfm = m >> 1;
    float prevd[kMD]; float prevna = 1.f;
    float usum[kMD];
#pragma unroll
    for (int d = 0; d < kMD; ++d) usum[d] = 0.f;
    float usq = 0.f;
    float path_len = 0.f, sum_dm2 = 0.f, max_dm = -1e30f;
    float first_sum = 0.f, second_sum = 0.f;
    float sum_cos = 0.f, sum_curv2 = 0.f;
    float max_curv = -1e30f, min_curv = 1e30f, min_cos = 1e30f;
    float ndir = 0.f;

    for (int i = 0; i < m; ++i) {
        float d[kMD]; float sq = 0.f;
#pragma unroll
        for (int dd = 0; dd < kMD; ++dd) {
            float v = C[(i + 1) * kMD + dd] - C[i * kMD + dd];
            d[dd] = v; sq += v * v;
        }
        float dm = (sq > 0.f) ? sqrtf(fmaxf(sq, 1e-30f)) : 0.f;
        path_len += dm; sum_dm2 += dm * dm;
        max_dm = fmaxf(max_dm, dm);
        if (i < halfm + 1) first_sum  += dm;
        if (i >= halfm)    second_sum += dm;

        float na  = fmaxf(dm, 1e-8f);
        float inv = 1.f / na;
#pragma unroll
        for (int dd = 0; dd < kMD; ++dd) {
            float u = d[dd] * inv;
            usum[dd] += u; usq += u * u;
        }
        if (i < 3) {   // first 3 deltas (m >= 3 always valid)
#pragma unroll
            for (int dd = 0; dd < kMD; ++dd) fout[21 + i * kMD + dd] = d[dd];
        }
        if (i > 0) {   // cos index i-1 < n-2 holds for all i < m
            float dot = 0.f;
#pragma unroll
            for (int dd = 0; dd < kMD; ++dd) dot += prevd[dd] * d[dd];
            float c    = dot / (prevna * na);
            float curv = 1.f - c;
            sum_cos += c; sum_curv2 += curv * curv;
            max_curv = fmaxf(max_curv, curv);
            min_curv = fminf(min_curv, curv);
            min_cos  = fminf(min_cos, c);
            if (c < 0.f) ndir += 1.f;
            if (i - 1 < 9) fout[51 + (i - 1)] = curv;
        }
#pragma unroll
        for (int dd = 0; dd < kMD; ++dd) prevd[dd] = d[dd];
        prevna = na;
    }

    const float ncf       = (float)(n - 2);
    const float mean_curv = (ncf - sum_cos) / ncf;             // sum(1-cos)/ncf
    const float curv_var  = (sum_curv2 - ncf * mean_curv * mean_curv)
                            / fmaxf(ncf - 1.f, 1.f);
    const float std_curv  = (ncf > 1.f) ? sqrtf(fmaxf(curv_var, 1e-30f)) : 0.f;
    const float mean_cos  = sum_cos / ncf;
    const float dir_chg   = ndir / fmaxf(ncf, 1.f);
    const float disp_ratio = total_disp / (path_len + 1e-9f);

    const float fh = first_sum  / (float)(halfm + 1);
    const float sh = second_sum / (float)(m - halfm);
    const float convergence = (fh - sh) / (fh + 1e-9f);
    const float cascade     = (sh - fh) / (fh + 1e-9f);

    float usum_sq = 0.f;
#pragma unroll
    for (int dd = 0; dd < kMD; ++dd) usum_sq += usum[dd] * usum[dd];
    const float npairs      = mf * (mf - 1.f) * 0.5f;
    // sum_{i<j} <u_i,u_j> = (|sum u|^2 - sum |u|^2) / 2
    const float parallelism = 0.5f * (usum_sq - usq) / fmaxf(npairs, 1.f);

    const float mean_dm = path_len / mf;
    const float dm_var  = (sum_dm2 - mf * mean_dm * mean_dm) / fmaxf(mf - 1.f, 1.f);
    const float std_dm  = (mf > 1.f) ? sqrtf(fmaxf(dm_var, 1e-30f)) : 0.f;
    const float jump    = (mean_dm > 1e-9f) ? (max_dm / mean_dm) : 1.f;

    const float cnt        = nf * (float)kMD;
    const float mean_coord = csum / cnt;
    const float coord_var  = (csumsq - cnt * mean_coord * mean_coord) / (cnt - 1.f);
    const float std_coord  = sqrtf(fmaxf(coord_var, 1e-30f));

    fout[0]  = total_disp;          fout[1]  = path_len;
    fout[2]  = disp_ratio;          fout[3]  = nf * 0.1f;
    fout[4]  = mean_curv;           fout[5]  = max_curv;
    fout[6]  = std_curv;            fout[7]  = max_curv - min_curv;
    fout[8]  = mean_cos;            fout[9]  = min_cos;
    fout[10] = dir_chg;             fout[11] = disp_ratio;       // linearity
    fout[12] = 1.f - disp_ratio;    fout[13] = convergence;      // loop_score
    fout[14] = parallelism;         fout[15] = jump;
    fout[16] = cascade;             fout[17] = mean_dm;
    fout[18] = std_dm;              fout[19] = mean_coord;
    fout[20] = std_coord;
}

// ---------------------------------------------------------------------------
// Kernel 2: feats[B,64] @ Wpad[64,48] via V_WMMA_F32_16X16X4_F32,
// then bias + LayerNorm (shfl_xor reduction inside 16-lane halves) + GELU.
// 256 threads = 8 waves, each wave owns 16 rows and all 3 N-tiles of 16.
// ---------------------------------------------------------------------------
__global__ __launch_bounds__(256) void gemm_ln_gelu_kernel(
    const float* __restrict__ feats, const float* __restrict__ W,
    const float* __restrict__ bias,  const float* __restrict__ lnw,
    const float* __restrict__ lnb,   float* __restrict__ out)
{
    __shared__ float Wlds[kF * kH];   // 64 x 48, rows 60..63 zero
    const int tid = threadIdx.x;
    for (int idx = tid; idx < kF * kH; idx += 256) {
        int row = idx / kH;
        Wlds[idx] = (row < 60) ? W[idx] : 0.f;
    }
    __syncthreads();

    const int wave    = tid >> 5;
    const int lane    = tid & 31;
    const int halfsel = lane >> 4;     // 0: lanes 0-15, 1: lanes 16-31
    const int l16     = lane & 15;
    const int rowBase = blockIdx.x * 128 + wave * 16;

    // A fragment base: row (rowBase+l16), K offset halfsel*2 per ISA 16x4 layout
    const float* arow = feats + (size_t)(rowBase + l16) * kF + halfsel * 2;

    v8f acc0 = {}; v8f acc1 = {}; v8f acc2 = {};
#pragma unroll
    for (int kk = 0; kk < 16; ++kk) {
        const int k0 = kk * 4 + halfsel * 2;   // B K-rows held by this lane-half
        v2f a; a.x = arow[kk * 4]; a.y = arow[kk * 4 + 1];
        const float* wr0 = &Wlds[k0 * kH];
        const float* wr1 = wr0 + kH;
        v2f b0, b1, b2;
        b0.x = wr0[l16];      b0.y = wr1[l16];
        b1.x = wr0[l16 + 16]; b1.y = wr1[l16 + 16];
        b2.x = wr0[l16 + 32]; b2.y = wr1[l16 + 32];
        acc0 = __builtin_amdgcn_wmma_f32_16x16x4_f32(false, a, false, b0,
                                                     (short)0, acc0, false, false);
        acc1 = __builtin_amdgcn_wmma_f32_16x16x4_f32(false, a, false, b1,
                                                     (short)0, acc1, false, false);
        acc2 = __builtin_amdgcn_wmma_f32_16x16x4_f32(false, a, false, b2,
                                                     (short)0, acc2, false, false);
    }

    const float b0c = bias[l16], b1c = bias[l16 + 16], b2c = bias[l16 + 32];
    const float w0c = lnw[l16],  w1c = lnw[l16 + 16],  w2c = lnw[l16 + 32];
    const float g0c = lnb[l16],  g1c = lnb[l16 + 16],  g2c = lnb[l16 + 32];
    const float kInvSqrt2 = 0.70710678118654752440f;

    // C/D layout: VGPR r holds row (r + halfsel*8), column l16 (+16 per tile)
#pragma unroll
    for (int r = 0; r < 8; ++r) {
        float h0 = acc0[r] + b0c;
        float h1 = acc1[r] + b1c;
        float h2 = acc2[r] + b2c;
        float s  = h0 + h1 + h2;
        float sq = h0 * h0 + h1 * h1 + h2 * h2;
#pragma unroll
        for (int msk = 1; msk < 16; msk <<= 1) {   // stays within 16-lane half
            s  += __shfl_xor(s,  msk, 32);
            sq += __shfl_xor(sq, msk, 32);
        }
        const float mu  = s * (1.f / 48.f);
        const float var = sq * (1.f / 48.f) - mu * mu;
        const float inv = rsqrtf(var + 1e-5f);
        const int   row = rowBase + halfsel * 8 + r;
        float x0 = (h0 - mu) * inv * w0c + g0c;
        float x1 = (h1 - mu) * inv * w1c + g1c;
        float x2 = (h2 - mu) * inv * w2c + g2c;
        out[(size_t)row * kH + l16]      = 0.5f * x0 * (1.f + erff(x0 * kInvSqrt2));
        out[(size_t)row * kH + l16 + 16] = 0.5f * x1 * (1.f + erff(x1 * kInvSqrt2));
        out[(size_t)row * kH + l16 + 32] = 0.5f * x2 * (1.f + erff(x2 * kInvSqrt2));
    }
}

// ---------------------------------------------------------------------------
extern "C" void kernel_launch(void* const* d_in, const int* in_sizes, int n_in,
                              void* d_out, int out_size, void* d_ws, size_t ws_size,
                              hipStream_t stream)
{
    (void)n_in; (void)out_size; (void)ws_size;
    const float* coords  = (const float*)d_in[0];
    const int*   lengths = (const int*)  d_in[1];
    const float* W       = (const float*)d_in[2];
    const float* bias    = (const float*)d_in[3];
    const float* lnw     = (const float*)d_in[4];
    const float* lnb     = (const float*)d_in[5];
    float*       out     = (float*)d_out;
    float*       feats   = (float*)d_ws;           // B x 64 f32 scratch

    const int B = in_sizes[0] / kSMD;              // 32768

    traj_features_kernel<<<B / 32, 32, 0, stream>>>(coords, lengths, feats);
    gemm_ln_gelu_kernel<<<B / 128, 256, 0, stream>>>(feats, W, bias, lnw, lnb, out);
}